// Net_22488448762768
// MI455X (gfx1250) — compile-verified
//
#include <hip/hip_runtime.h>

#define N_NODES   200000
#define N_EDGES   6400000
#define N_GRAPHS  2000

typedef float v2f __attribute__((ext_vector_type(2)));
typedef float v8f __attribute__((ext_vector_type(8)));

// ---------------------------------------------------------------------------
// WMMA helpers: D = A(16x4 f32) * B(4x16 f32) + C(16x16 f32), wave32.
// A layout: lanes 0-15 -> M=lane, vgpr{0,1} = K{0,1}; lanes 16-31 -> K{2,3}.
// B layout: lanes 0-15 -> N=lane, vgpr{0,1} = K{0,1}; lanes 16-31 -> K{2,3}.
// C/D layout: lane -> N=lane&15, vgpr r -> M = r + 8*(lane>>4).
// ---------------------------------------------------------------------------
__device__ __forceinline__ v8f wmma_k4(v2f a, v2f b, v8f c) {
  return __builtin_amdgcn_wmma_f32_16x16x4_f32(false, a, false, b, (short)0, c,
                                               false, false);
}

__device__ __forceinline__ v8f relu8(v8f d) {
#pragma unroll
  for (int i = 0; i < 8; ++i) d[i] = fmaxf(d[i], 0.0f);
  return d;
}

// 16x16 (tile, row stride 17) @ 16x16 W (row-major [K][N]) + bias[N]
__device__ __forceinline__ v8f mm16_from_tile(const float* tile,
                                              const float* __restrict__ W,
                                              const float* __restrict__ bias,
                                              int lane) {
  const int n  = lane & 15;   // = M for A fragment, = N for B/C fragments
  const int hl = lane >> 4;
  const float bn = bias[n];
  v8f c;
#pragma unroll
  for (int i = 0; i < 8; ++i) c[i] = bn;
#pragma unroll
  for (int kk = 0; kk < 16; kk += 4) {
    const int k0 = kk + 2 * hl;
    v2f a, b;
    a.x = tile[n * 17 + k0];
    a.y = tile[n * 17 + k0 + 1];
    b.x = W[(k0)     * 16 + n];
    b.y = W[(k0 + 1) * 16 + n];
    c = wmma_k4(a, b, c);
  }
  return c;
}

// ---------------------------------------------------------------------------
// Utility kernels
// ---------------------------------------------------------------------------
__global__ void zero_kernel(float* __restrict__ p, int n) {
  int i = blockIdx.x * blockDim.x + threadIdx.x;
  if (i < n) p[i] = 0.0f;
}

__global__ void scatter2_kernel(const float* __restrict__ x,
                                const int* __restrict__ src,
                                const int* __restrict__ dst,
                                float* __restrict__ aggr) {
  int e = blockIdx.x * blockDim.x + threadIdx.x;
  if (e >= N_EDGES) return;
  int s = src[e], d = dst[e];
  unsafeAtomicAdd(aggr + d * 2 + 0, x[s * 2 + 0]);
  unsafeAtomicAdd(aggr + d * 2 + 1, x[s * 2 + 1]);
}

__global__ void scatter16_kernel(const float* __restrict__ h,
                                 const int* __restrict__ src,
                                 const int* __restrict__ dst,
                                 float* __restrict__ aggr) {
  int e = blockIdx.x * blockDim.x + threadIdx.x;
  if (e >= N_EDGES) return;
  int s = src[e], d = dst[e];
  const float4* hp = (const float4*)(h + (size_t)s * 16);
  float* ap = aggr + (size_t)d * 16;
#pragma unroll
  for (int q = 0; q < 4; ++q) {
    float4 v = hp[q];
    unsafeAtomicAdd(ap + 4 * q + 0, v.x);
    unsafeAtomicAdd(ap + 4 * q + 1, v.y);
    unsafeAtomicAdd(ap + 4 * q + 2, v.z);
    unsafeAtomicAdd(ap + 4 * q + 3, v.w);
  }
}

__global__ void pool_kernel(const float* __restrict__ h,
                            const int* __restrict__ batch,
                            float* __restrict__ g) {
  int v = blockIdx.x * blockDim.x + threadIdx.x;
  if (v >= N_NODES) return;
  int b = batch[v];
  const float4* hp = (const float4*)(h + (size_t)v * 16);
  float* gp = g + (size_t)b * 16;
#pragma unroll
  for (int q = 0; q < 4; ++q) {
    float4 x = hp[q];
    unsafeAtomicAdd(gp + 4 * q + 0, x.x);
    unsafeAtomicAdd(gp + 4 * q + 1, x.y);
    unsafeAtomicAdd(gp + 4 * q + 2, x.z);
    unsafeAtomicAdd(gp + 4 * q + 3, x.w);
  }
}

// ---------------------------------------------------------------------------
// GIN layer (16-d input):  out = relu( relu((in+aggr)@Wa + ba) @ Wb + bb )
// 4 waves / block, one 16-node tile per wave. 3125 blocks -> 12500 tiles.
// ---------------------------------------------------------------------------
#define LWAVES 4

__global__ void layer16_kernel(const float* __restrict__ in,
                               const float* __restrict__ aggr,
                               const float* __restrict__ Wa,
                               const float* __restrict__ ba,
                               const float* __restrict__ Wb,
                               const float* __restrict__ bb,
                               float* __restrict__ out) {
  __shared__ float tiles[LWAVES][16 * 17];
  const int wave = threadIdx.x >> 5;
  const int lane = threadIdx.x & 31;
  const int t    = blockIdx.x * LWAVES + wave;
  float* tile = tiles[wave];

  // Load (in + aggr) 16x16 tile into LDS: each lane handles 8 floats.
  {
    const int row  = lane & 15;
    const int ch   = lane >> 4;
    const int node = t * 16 + row;
    const float4* ip = (const float4*)(in   + (size_t)node * 16 + ch * 8);
    const float4* ap = (const float4*)(aggr + (size_t)node * 16 + ch * 8);
    float4 v0 = ip[0], v1 = ip[1];
    float4 a0 = ap[0], a1 = ap[1];
    float* tp = tile + row * 17 + ch * 8;
    tp[0] = v0.x + a0.x; tp[1] = v0.y + a0.y; tp[2] = v0.z + a0.z; tp[3] = v0.w + a0.w;
    tp[4] = v1.x + a1.x; tp[5] = v1.y + a1.y; tp[6] = v1.z + a1.z; tp[7] = v1.w + a1.w;
  }
  __syncthreads();

  v8f d = relu8(mm16_from_tile(tile, Wa, ba, lane));
  __syncthreads();
  {
    const int n = lane & 15, hl = lane >> 4;
#pragma unroll
    for (int r = 0; r < 8; ++r) tile[(r + 8 * hl) * 17 + n] = d[r];
  }
  __syncthreads();
  v8f d2 = relu8(mm16_from_tile(tile, Wb, bb, lane));
  {
    const int n = lane & 15, hl = lane >> 4;
#pragma unroll
    for (int r = 0; r < 8; ++r)
      out[(size_t)(t * 16 + r + 8 * hl) * 16 + n] = d2[r];
  }
}

// ---------------------------------------------------------------------------
// GIN layer 1 (2-d input): single K=4 WMMA with K=2,3 zero-padded.
// ---------------------------------------------------------------------------
__global__ void layer1_kernel(const float* __restrict__ x,
                              const float* __restrict__ aggr2,
                              const float* __restrict__ W1a,
                              const float* __restrict__ b1a,
                              const float* __restrict__ W1b,
                              const float* __restrict__ b1b,
                              float* __restrict__ out) {
  __shared__ float tiles[LWAVES][16 * 17];
  const int wave = threadIdx.x >> 5;
  const int lane = threadIdx.x & 31;
  const int t    = blockIdx.x * LWAVES + wave;
  float* tile = tiles[wave];
  const int n  = lane & 15;
  const int hl = lane >> 4;

  v2f a = {0.0f, 0.0f};
  v2f b = {0.0f, 0.0f};
  if (hl == 0) {                      // K=0,1 live in lanes 0-15 only
    const int node = t * 16 + n;
    a.x = x[(size_t)node * 2 + 0] + aggr2[(size_t)node * 2 + 0];
    a.y = x[(size_t)node * 2 + 1] + aggr2[(size_t)node * 2 + 1];
    b.x = W1a[0 * 16 + n];
    b.y = W1a[1 * 16 + n];
  }
  const float bn = b1a[n];
  v8f c;
#pragma unroll
  for (int i = 0; i < 8; ++i) c[i] = bn;
  c = relu8(wmma_k4(a, b, c));

#pragma unroll
  for (int r = 0; r < 8; ++r) tile[(r + 8 * hl) * 17 + n] = c[r];
  __syncthreads();
  v8f d2 = relu8(mm16_from_tile(tile, W1b, b1b, lane));
#pragma unroll
  for (int r = 0; r < 8; ++r)
    out[(size_t)(t * 16 + r + 8 * hl) * 16 + n] = d2[r];
}

// ---------------------------------------------------------------------------
// Head: out = relu(g@Wl1 + bl1) @ Wl2 + bl2 ; one wave per 16-graph tile.
// ---------------------------------------------------------------------------
__global__ void head_kernel(const float* __restrict__ g,
                            const float* __restrict__ Wl1,
                            const float* __restrict__ bl1,
                            const float* __restrict__ Wl2,
                            const float* __restrict__ bl2,
                            float* __restrict__ out) {
  __shared__ float tile[16 * 17];
  const int lane = threadIdx.x & 31;
  const int t    = blockIdx.x;
  {
    const int row = lane & 15;
    const int ch  = lane >> 4;
    const int gr  = t * 16 + row;
    const float4* gp = (const float4*)(g + (size_t)gr * 16 + ch * 8);
    float4 v0 = gp[0], v1 = gp[1];
    float* tp = tile + row * 17 + ch * 8;
    tp[0] = v0.x; tp[1] = v0.y; tp[2] = v0.z; tp[3] = v0.w;
    tp[4] = v1.x; tp[5] = v1.y; tp[6] = v1.z; tp[7] = v1.w;
  }
  __syncthreads();
  v8f d = relu8(mm16_from_tile(tile, Wl1, bl1, lane));
  __syncthreads();
  {
    const int n = lane & 15, hl = lane >> 4;
#pragma unroll
    for (int r = 0; r < 8; ++r) tile[(r + 8 * hl) * 17 + n] = d[r];
  }
  __syncthreads();
  if (lane < 16) {
    float acc = bl2[0];
#pragma unroll
    for (int k = 0; k < 16; ++k) acc += tile[lane * 17 + k] * Wl2[k];
    out[t * 16 + lane] = acc;
  }
}

// ---------------------------------------------------------------------------
extern "C" void kernel_launch(void* const* d_in, const int* in_sizes, int n_in,
                              void* d_out, int out_size, void* d_ws, size_t ws_size,
                              hipStream_t stream) {
  const float* x     = (const float*)d_in[0];
  const int*   edge  = (const int*)  d_in[1];
  const int*   batch = (const int*)  d_in[2];
  const float* W1a = (const float*)d_in[3];  const float* b1a = (const float*)d_in[4];
  const float* W1b = (const float*)d_in[5];  const float* b1b = (const float*)d_in[6];
  const float* W2a = (const float*)d_in[7];  const float* b2a = (const float*)d_in[8];
  const float* W2b = (const float*)d_in[9];  const float* b2b = (const float*)d_in[10];
  const float* W3a = (const float*)d_in[11]; const float* b3a = (const float*)d_in[12];
  const float* W3b = (const float*)d_in[13]; const float* b3b = (const float*)d_in[14];
  const float* Wl1 = (const float*)d_in[15]; const float* bl1 = (const float*)d_in[16];
  const float* Wl2 = (const float*)d_in[17]; const float* bl2 = (const float*)d_in[18];

  const int* src = edge;
  const int* dst = edge + N_EDGES;

  float* ws   = (float*)d_ws;
  float* aggr = ws;                       // N_NODES*16 floats (layer1 uses first N*2)
  float* hA   = aggr + (size_t)N_NODES * 16;
  float* hB   = hA   + (size_t)N_NODES * 16;
  float* g    = hB   + (size_t)N_NODES * 16;  // N_GRAPHS*16 floats
  float* out  = (float*)d_out;

  const int EB = (N_EDGES + 255) / 256;       // 25000
  const int TB = (N_NODES / 16) / LWAVES;     // 3125 blocks, 12500 tiles exact
  const int GB = N_GRAPHS / 16;               // 125 tiles exact

  // ---- Layer 1 (d=2) ----
  zero_kernel<<<(N_NODES * 2 + 255) / 256, 256, 0, stream>>>(aggr, N_NODES * 2);
  scatter2_kernel<<<EB, 256, 0, stream>>>(x, src, dst, aggr);
  layer1_kernel<<<TB, 32 * LWAVES, 0, stream>>>(x, aggr, W1a, b1a, W1b, b1b, hA);

  // ---- Layer 2 ----
  zero_kernel<<<(N_NODES * 16 + 255) / 256, 256, 0, stream>>>(aggr, N_NODES * 16);
  scatter16_kernel<<<EB, 256, 0, stream>>>(hA, src, dst, aggr);
  layer16_kernel<<<TB, 32 * LWAVES, 0, stream>>>(hA, aggr, W2a, b2a, W2b, b2b, hB);

  // ---- Layer 3 ----
  zero_kernel<<<(N_NODES * 16 + 255) / 256, 256, 0, stream>>>(aggr, N_NODES * 16);
  scatter16_kernel<<<EB, 256, 0, stream>>>(hB, src, dst, aggr);
  layer16_kernel<<<TB, 32 * LWAVES, 0, stream>>>(hB, aggr, W3a, b3a, W3b, b3b, hA);

  // ---- Pool + head ----
  zero_kernel<<<(N_GRAPHS * 16 + 255) / 256, 256, 0, stream>>>(g, N_GRAPHS * 16);
  pool_kernel<<<(N_NODES + 255) / 256, 256, 0, stream>>>(hA, batch, g);
  head_kernel<<<GB, 32, 0, stream>>>(g, Wl1, bl1, Wl2, bl2, out);
}